// hdc_classifier_11759620456581
// MI455X (gfx1250) — compile-verified
//
#include <hip/hip_runtime.h>
#include <hip/hip_bf16.h>

// Problem dims (fixed by the reference)
#define D_DIM 4096   // hypervector dimension
#define P_DIM 1024   // positions (32*32)
#define B_DIM 64     // batch
#define L_DIM 256    // levels
#define C_DIM 100    // classes

typedef __attribute__((ext_vector_type(2))) float v2f;
typedef __attribute__((ext_vector_type(8))) float v8f;

// ---------------------------------------------------------------------------
// Kernel A: bind + multiset + hard_quantize.
// Grid: (D/256, B/8), block 256 threads (8 wave32).
// Each block owns a 256-wide d-slice (one d per thread) and 8 batches.
// Level indices for its 8 batches are computed once into LDS (32 KB) and
// broadcast-read in the inner loop; pos[p, d-slice] is loaded once per p and
// reused across all 8 batches. vw gathers are coalesced across threads
// (consecutive d) and L2-resident (vw = 4 MB << 192 MB L2).
// All math is exact in f32 (sums of +/-1), bit-identical to the reference.
// ---------------------------------------------------------------------------
__global__ __launch_bounds__(256) void hdc_encode_kernel(
    const float* __restrict__ x,    // [B, P]
    const float* __restrict__ pos,  // [P, D]
    const float* __restrict__ vw,   // [L, D]
    float* __restrict__ enc)        // [B, D] (workspace)
{
  __shared__ int sidx[8 * P_DIM];   // 32 KB

  const int t  = threadIdx.x;
  const int d  = blockIdx.x * 256 + t;
  const int bg = blockIdx.y * 8;

  // Quantize x -> level index for this block's 8 batches (round half-to-even).
  for (int i = t; i < 8 * P_DIM; i += 256) {
    const int b = i >> 10;
    const int p = i & (P_DIM - 1);
    const float xv = x[(bg + b) * P_DIM + p];
    int l = (int)rintf(xv * 255.0f);
    l = l < 0 ? 0 : (l > (L_DIM - 1) ? (L_DIM - 1) : l);
    sidx[i] = l;
  }
  __syncthreads();

  float acc[8];
#pragma unroll
  for (int g = 0; g < 8; ++g) acc[g] = 0.0f;

  for (int p = 0; p < P_DIM; ++p) {
    const float pv = pos[p * D_DIM + d];
    // emits global_prefetch_b8 for the next pos rows
    const int pf = (p + 4 < P_DIM) ? (p + 4) : p;
    __builtin_prefetch(&pos[pf * D_DIM + d], 0, 0);
#pragma unroll
    for (int g = 0; g < 8; ++g) {
      const int l = sidx[g * P_DIM + p];          // uniform -> LDS broadcast
      acc[g] = fmaf(pv, vw[l * D_DIM + d], acc[g]);
    }
  }

#pragma unroll
  for (int g = 0; g < 8; ++g)
    enc[(bg + g) * D_DIM + d] = acc[g] > 0.0f ? 1.0f : -1.0f;
}

// ---------------------------------------------------------------------------
// Kernel B: logits = enc @ W^T via V_WMMA_F32_16X16X4_F32 (exact f32 path).
// One wave per 16x16 output tile; K loop over D in steps of 4.
// A layout (16x4 f32, 2 VGPRs): lanes 0-15 -> M=lane, K={k,k+1};
//                               lanes 16-31 -> M=lane-16, K={k+2,k+3}.
// B layout (4x16, 2 VGPRs):     lanes 0-15 -> N=lane, K={k,k+1};
//                               lanes 16-31 -> N=lane-16, K={k+2,k+3}.
// Padded class columns (n >= 100) are zeroed by value (cndmask), keeping
// EXEC all-ones as WMMA requires; their D columns are never stored.
// ---------------------------------------------------------------------------
__global__ __launch_bounds__(32) void hdc_classify_wmma_kernel(
    const float* __restrict__ enc,  // [B, D]
    const float* __restrict__ W,    // [C, D]
    float* __restrict__ out)        // [B, C]
{
  const int lane = threadIdx.x;
  const int l15  = lane & 15;
  const int half = lane >> 4;       // 0: K pair {k,k+1}; 1: {k+2,k+3}
  const int m0   = blockIdx.x * 16; // batch tile
  const int n0   = blockIdx.y * 16; // class tile

  const int   n     = n0 + l15;
  const int   nclmp = (n < C_DIM) ? n : (C_DIM - 1);
  const float nmask = (n < C_DIM) ? 1.0f : 0.0f;

  const float* __restrict__ arow = enc + (m0 + l15) * D_DIM;
  const float* __restrict__ wrow = W   + nclmp * D_DIM;

  v8f c = {0.f, 0.f, 0.f, 0.f, 0.f, 0.f, 0.f, 0.f};

  for (int k = 0; k < D_DIM; k += 4) {
    const int ka = k + half * 2;
    v2f a, b;
    a.x = arow[ka];
    a.y = arow[ka + 1];
    b.x = wrow[ka] * nmask;
    b.y = wrow[ka + 1] * nmask;
    // 8 args: (neg_a, A, neg_b, B, c_mod, C, reuse_a, reuse_b)
    c = __builtin_amdgcn_wmma_f32_16x16x4_f32(
        /*neg_a=*/false, a, /*neg_b=*/false, b,
        /*c_mod=*/(short)0, c, /*reuse_a=*/false, /*reuse_b=*/false);
  }

  // D layout: VGPR i -> lanes 0-15: (M=i, N=lane); lanes 16-31: (M=i+8, N=lane-16)
#pragma unroll
  for (int i = 0; i < 8; ++i) {
    const int m = m0 + i + half * 8;
    if (n < C_DIM) out[m * C_DIM + n] = c[i];
  }
}

// ---------------------------------------------------------------------------
// Host launcher. Inputs: x[64*1024], pos[1024*4096], vw[256*4096], W[100*4096]
// Workspace: enc = 64*4096 floats (1 MB).
// ---------------------------------------------------------------------------
extern "C" void kernel_launch(void* const* d_in, const int* in_sizes, int n_in,
                              void* d_out, int out_size, void* d_ws, size_t ws_size,
                              hipStream_t stream) {
  const float* x   = (const float*)d_in[0];
  const float* pos = (const float*)d_in[1];
  const float* vw  = (const float*)d_in[2];
  const float* W   = (const float*)d_in[3];
  float* out = (float*)d_out;
  float* enc = (float*)d_ws;   // [B_DIM * D_DIM] floats

  dim3 gridA(D_DIM / 256, B_DIM / 8);   // 16 x 8 = 128 blocks
  hdc_encode_kernel<<<gridA, dim3(256), 0, stream>>>(x, pos, vw, enc);

  dim3 gridB(B_DIM / 16, (C_DIM + 15) / 16);  // 4 x 7 tiles
  hdc_classify_wmma_kernel<<<gridB, dim3(32), 0, stream>>>(enc, W, out);
}